// GetCMatrix_55482387530023
// MI455X (gfx1250) — compile-verified
//
#include <hip/hip_runtime.h>
#include <math.h>

typedef __attribute__((ext_vector_type(2))) float v2f;
typedef __attribute__((ext_vector_type(8))) float v8f;

// ---------------------------------------------------------------------------
// Kernel 1: assignments[i] = sum_d inputs[i,d] * (9 * 10^d)  via V_WMMA_F32_16X16X4_F32
// Each wave processes 16-row tiles. A-matrix 16x4 f32 layout (ISA 7.12.2):
//   lane L: M = L%16, VGPR0 = A[M, 4c + 2*(L/16)], VGPR1 = A[M, 4c + 2*(L/16) + 1]
// B 4x16 has w broadcast in every column, so B slot value depends only on K:
//   lane L: VGPR0 = w[4c + 2*(L/16)], VGPR1 = w[4c + 2*(L/16)+1]
// D 16x16: lane L, VGPR v holds D[M = v + 8*(L/16), N = L%16]; all columns equal,
// so lanes with L%16==0 (lanes 0 and 16) hold the 16 results across v=0..7.
// ---------------------------------------------------------------------------
__global__ void assign_wmma_kernel(const float* __restrict__ inputs,
                                   float* __restrict__ assignments,
                                   int num_tiles) {
    const int lane = threadIdx.x & 31;
    const int half = lane >> 4;           // 0 or 1
    const int m    = lane & 15;           // row within tile
    const int wavesPerBlock = blockDim.x >> 5;
    int wave   = blockIdx.x * wavesPerBlock + (threadIdx.x >> 5);
    int nWaves = gridDim.x * wavesPerBlock;

    // w[k] = 9 * 10^k (fold the x*9 scaling into the weight vector)
    float wv[16];
    float p = 9.0f;
#pragma unroll
    for (int k = 0; k < 16; ++k) { wv[k] = p; p *= 10.0f; }

    for (int t = wave; t < num_tiles; t += nWaves) {
        const float* tile = inputs + (size_t)t * 16 * 16;  // 16 rows x 16 cols, row-major
        v8f acc = {};
#pragma unroll
        for (int c = 0; c < 4; ++c) {
            const int k0 = 4 * c + 2 * half;
            v2f a = *(const v2f*)(tile + m * 16 + k0);     // 8B aligned
            v2f b;
            b.x = wv[k0];
            b.y = wv[k0 + 1];
            acc = __builtin_amdgcn_wmma_f32_16x16x4_f32(
                /*neg_a=*/false, a, /*neg_b=*/false, b,
                /*c_mod=*/(short)0, acc, /*reuse_a=*/false, /*reuse_b=*/false);
        }
        if (m == 0) {
#pragma unroll
            for (int v = 0; v < 8; ++v)
                assignments[(size_t)t * 16 + v + 8 * half] = acc[v];
        }
    }
}

// ---------------------------------------------------------------------------
// Scalar tail for N not divisible by 16 (not used for the reference shapes).
// ---------------------------------------------------------------------------
__global__ void assign_tail_kernel(const float* __restrict__ inputs,
                                   float* __restrict__ assignments,
                                   int start, int N) {
    int i = start + blockIdx.x * blockDim.x + threadIdx.x;
    if (i < N) {
        float acc = 0.0f, p = 9.0f;
#pragma unroll
        for (int d = 0; d < 16; ++d) { acc += inputs[(size_t)i * 16 + d] * p; p *= 10.0f; }
        assignments[i] = acc;
    }
}

// ---------------------------------------------------------------------------
// Kernel 2: single fused pass over edges.
//   degrees[col] += w;  As[row] += w * assignments[col];  scalars[1] += sum(w)
// assignments/degrees/As are 4MB each -> resident in the 192MB L2, so the
// random gather + atomics stay on-chip; HBM traffic is the 12B/edge stream.
// ---------------------------------------------------------------------------
__global__ void edge_kernel(const int* __restrict__ row,
                            const int* __restrict__ col,
                            const float* __restrict__ w,
                            const float* __restrict__ assign,
                            float* __restrict__ degrees,
                            float* __restrict__ As,
                            float* __restrict__ scalars,
                            int E) {
    __shared__ float red[256];
    const int stride = gridDim.x * blockDim.x;
    float wsum = 0.0f;
    for (int e = blockIdx.x * blockDim.x + threadIdx.x; e < E; e += stride) {
        // stream-ahead prefetch of the weight array (emits global_prefetch_b8)
        if (e + stride < E) __builtin_prefetch(w + e + stride, 0, 1);
        const int   r  = row[e];
        const int   c  = col[e];
        const float we = w[e];
        wsum += we;
        atomicAdd(&degrees[c], we);
        atomicAdd(&As[r], we * assign[c]);
    }
    red[threadIdx.x] = wsum;
    __syncthreads();
    for (int s = blockDim.x >> 1; s > 0; s >>= 1) {
        if (threadIdx.x < (unsigned)s) red[threadIdx.x] += red[threadIdx.x + s];
        __syncthreads();
    }
    if (threadIdx.x == 0) atomicAdd(&scalars[1], red[0]);
}

// ---------------------------------------------------------------------------
// Kernel 3: reductions over N:
//   scalars[0] += sum(assign)          (cluster_sizes)
//   scalars[2] += dot(As, assign)      (graph_pooled)
//   scalars[3] += dot(assign, degrees) (normalizer_left == normalizer_right)
// ---------------------------------------------------------------------------
__global__ void reduce_kernel(const float* __restrict__ assign,
                              const float* __restrict__ degrees,
                              const float* __restrict__ As,
                              float* __restrict__ scalars,
                              int N) {
    __shared__ float r0[256], r1[256], r2[256];
    const int stride = gridDim.x * blockDim.x;
    float sa = 0.0f, gp = 0.0f, nl = 0.0f;
    for (int i = blockIdx.x * blockDim.x + threadIdx.x; i < N; i += stride) {
        const float a = assign[i];
        sa += a;
        gp += As[i] * a;
        nl += a * degrees[i];
    }
    r0[threadIdx.x] = sa; r1[threadIdx.x] = gp; r2[threadIdx.x] = nl;
    __syncthreads();
    for (int s = blockDim.x >> 1; s > 0; s >>= 1) {
        if (threadIdx.x < (unsigned)s) {
            r0[threadIdx.x] += r0[threadIdx.x + s];
            r1[threadIdx.x] += r1[threadIdx.x + s];
            r2[threadIdx.x] += r2[threadIdx.x + s];
        }
        __syncthreads();
    }
    if (threadIdx.x == 0) {
        atomicAdd(&scalars[0], r0[0]);
        atomicAdd(&scalars[2], r1[0]);
        atomicAdd(&scalars[3], r2[0]);
    }
}

// ---------------------------------------------------------------------------
// Kernel 4: finalize both losses.
//   normalizer    = nl^2 / (2 * num_edges)
//   spectral_loss = -(graph_pooled - normalizer) / (2 * num_edges)
//   collapse_loss = 0.1 * (|sum_assign| / N * sqrt(10) - 1)
// ---------------------------------------------------------------------------
__global__ void finalize_kernel(const float* __restrict__ scalars,
                                float* __restrict__ out, int N) {
    if (blockIdx.x == 0 && threadIdx.x == 0) {
        const float sa   = scalars[0];
        const float ne   = scalars[1];
        const float gp   = scalars[2];
        const float nl   = scalars[3];
        const float twoE = 2.0f * ne;
        const float normalizer = nl * nl / twoE;
        out[N]     = -(gp - normalizer) / twoE;
        out[N + 1] = 0.1f * (fabsf(sa) / (float)N * sqrtf(10.0f) - 1.0f);
    }
}

extern "C" void kernel_launch(void* const* d_in, const int* in_sizes, int n_in,
                              void* d_out, int out_size, void* d_ws, size_t ws_size,
                              hipStream_t stream) {
    const float* inputs      = (const float*)d_in[0];   // [N,16] f32
    const int*   edge_index  = (const int*)d_in[1];     // [2,E] i32 (row then col)
    const float* edge_weight = (const float*)d_in[2];   // [E] f32

    const int N = in_sizes[0] / 16;
    const int E = in_sizes[2];

    float* out     = (float*)d_out;       // [0..N-1]=assignments, [N]=spectral, [N+1]=collapse
    float* degrees = (float*)d_ws;        // N floats
    float* As      = degrees + N;         // N floats
    float* scalars = As + N;              // 4 floats: sumA, sumW, graph_pooled, norm_left

    // zero scratch (captured as a memset node)
    hipMemsetAsync(d_ws, 0, (size_t)(2 * (size_t)N + 4) * sizeof(float), stream);

    const int tiles = N / 16;
    const int rem_start = tiles * 16;

    if (tiles > 0)
        assign_wmma_kernel<<<2048, 256, 0, stream>>>(inputs, out, tiles);
    if (rem_start < N)
        assign_tail_kernel<<<(N - rem_start + 255) / 256, 256, 0, stream>>>(
            inputs, out, rem_start, N);

    edge_kernel<<<8192, 256, 0, stream>>>(edge_index, edge_index + E, edge_weight,
                                          out, degrees, As, scalars, E);

    reduce_kernel<<<2048, 256, 0, stream>>>(out, degrees, As, scalars, N);

    finalize_kernel<<<1, 64, 0, stream>>>(scalars, out, N);
}